// ST_BLOCK_2_4337916969699
// MI455X (gfx1250) — compile-verified
//
#include <hip/hip_runtime.h>
#include <math.h>

#define B_ 64
#define CI 64
#define CO 64
#define N_ 207
#define T_ 24
#define NT 4968       /* N_*T_ */
#define PB 317952     /* CO*NT */
#define S1 20348928   /* B_*PB */
#define SLOPE 0.01f
#define EPS_BN 1e-5f
#define EPS_LN 1e-5f

typedef __attribute__((ext_vector_type(2))) float v2f;
typedef __attribute__((ext_vector_type(8))) float v8f;

__device__ __forceinline__ v8f wmma4(v2f a, v2f b, v8f c) {
  // D = A(16x4,f32) * B(4x16,f32) + C(16x16,f32)
  return __builtin_amdgcn_wmma_f32_16x16x4_f32(false, a, false, b, (short)0, c,
                                               false, false);
}
__device__ __forceinline__ float lrelu(float v) { return v >= 0.f ? v : SLOPE * v; }
__device__ __forceinline__ float sigm(float v) { return 1.f / (1.f + __expf(-v)); }

// ---------------- K1: temporal conv (1x3, pad 1) + bias + lrelu ----------------
// GEMM per (b,n): M=64 (o), K=192 (c*3+tap), cols=24 (t, padded to 32)
__global__ __launch_bounds__(256) void k_conv(const float* __restrict__ x,
                                              const float* __restrict__ Wt,
                                              const float* __restrict__ bt,
                                              float* __restrict__ xin1) {
  int b = blockIdx.x / N_, n = blockIdx.x % N_;
  __shared__ float Bm[192 * 32];
  for (int idx = threadIdx.x; idx < 192 * 32; idx += 256) {
    int k = idx >> 5, t = idx & 31, c = k / 3, tap = k % 3, tt = t + tap - 1;
    float v = 0.f;
    if (t < T_ && tt >= 0 && tt < T_) v = x[((b * CI + c) * N_ + n) * T_ + tt];
    Bm[idx] = v;
  }
  __syncthreads();
  int lane = threadIdx.x & 31, wave = threadIdx.x >> 5;
  int lrow = lane & 15, khalf = (lane >> 4) * 2, crow = (lane >> 4) * 8;
  int m0 = (wave >> 1) * 16, n0 = (wave & 1) * 16;
  v8f acc = {};
  for (int k0 = 0; k0 < 192; k0 += 4) {
    v2f a, bb;
    a.x = Wt[(m0 + lrow) * 192 + k0 + khalf];
    a.y = Wt[(m0 + lrow) * 192 + k0 + khalf + 1];
    bb.x = Bm[(k0 + khalf) * 32 + n0 + lrow];
    bb.y = Bm[(k0 + khalf + 1) * 32 + n0 + lrow];
    acc = wmma4(a, bb, acc);
  }
  for (int r = 0; r < 8; r++) {
    int row = m0 + crow + r, col = n0 + lrow;
    if (col < T_)
      xin1[((b * CO + row) * N_ + n) * T_ + col] = lrelu(acc[r] + bt[row]);
  }
}

// ---------------- K2/K4: channel mix  out[b,o,col] = W[o,c] in[b,c,col] + bias --
__global__ __launch_bounds__(256) void k_chanmix(const float* __restrict__ in,
                                                 const float* __restrict__ W,
                                                 const float* __restrict__ bias,
                                                 float* __restrict__ out, int Mout) {
  int b = blockIdx.y;
  int col0 = blockIdx.x * 32;
  __shared__ float Bm[64 * 32];
  for (int idx = threadIdx.x; idx < 64 * 32; idx += 256) {
    int k = idx >> 5, t = idx & 31, col = col0 + t;
    Bm[idx] = (col < NT) ? in[(b * 64 + k) * NT + col] : 0.f;
  }
  __syncthreads();
  int lane = threadIdx.x & 31, wave = threadIdx.x >> 5;
  int lrow = lane & 15, khalf = (lane >> 4) * 2, crow = (lane >> 4) * 8;
  int ntiles = (Mout >> 4) * 2;
  for (int tau = wave; tau < ntiles; tau += 8) {
    int m0 = (tau >> 1) * 16, n0 = (tau & 1) * 16;
    v8f acc = {};
    for (int k0 = 0; k0 < 64; k0 += 4) {
      v2f a, bb;
      a.x = W[(m0 + lrow) * 64 + k0 + khalf];
      a.y = W[(m0 + lrow) * 64 + k0 + khalf + 1];
      bb.x = Bm[(k0 + khalf) * 32 + n0 + lrow];
      bb.y = Bm[(k0 + khalf + 1) * 32 + n0 + lrow];
      acc = wmma4(a, bb, acc);
    }
    for (int r = 0; r < 8; r++) {
      int row = m0 + crow + r, col = col0 + n0 + lrow;
      if (col < NT) out[(b * Mout + row) * NT + col] = acc[r] + bias[row];
    }
  }
}

// ---------------- K3: out[b,c,n,t] = sum_m adj[n,m] in[b,c,m,t] ----------------
__global__ __launch_bounds__(256) void k_adjmm(const float* __restrict__ adj,
                                               const float* __restrict__ in,
                                               float* __restrict__ out) {
  int b = blockIdx.x >> 6, c = blockIdx.x & 63;
  __shared__ float Bm[208 * 32];
  const float* src = in + (b * 64 + c) * NT;
  for (int idx = threadIdx.x; idx < 208 * 32; idx += 256) {
    int k = idx >> 5, t = idx & 31;
    Bm[idx] = (k < N_ && t < T_) ? src[k * T_ + t] : 0.f;
  }
  __syncthreads();
  int lane = threadIdx.x & 31, wave = threadIdx.x >> 5;
  int lrow = lane & 15, khalf = (lane >> 4) * 2, crow = (lane >> 4) * 8;
  for (int tau = wave; tau < 26; tau += 8) {
    int m0 = (tau >> 1) * 16, n0 = (tau & 1) * 16;
    v8f acc = {};
    for (int k0 = 0; k0 < 208; k0 += 4) {
      int nrow = m0 + lrow, ka = k0 + khalf, kb = ka + 1;
      v2f a, bb;
      a.x = (nrow < N_ && ka < N_) ? adj[nrow * N_ + ka] : 0.f;
      a.y = (nrow < N_ && kb < N_) ? adj[nrow * N_ + kb] : 0.f;
      bb.x = Bm[ka * 32 + n0 + lrow];
      bb.y = Bm[kb * 32 + n0 + lrow];
      acc = wmma4(a, bb, acc);
    }
    for (int r = 0; r < 8; r++) {
      int row = m0 + crow + r, col = n0 + lrow;
      if (row < N_ && col < T_)
        out[(b * 64 + c) * NT + row * T_ + col] = acc[r];
    }
  }
}

// ------- K5: adjacency GEMM on filt & gate channels + GLU gating (fused) -------
__global__ __launch_bounds__(256) void k_adjmm_gated(const float* __restrict__ adj,
                                                     const float* __restrict__ h2,
                                                     float* __restrict__ x1) {
  int b = blockIdx.x >> 6, c = blockIdx.x & 63;
  __shared__ float Bf[208 * 32];
  __shared__ float Bg[208 * 32];
  const float* sf = h2 + (b * 128 + c) * NT;
  const float* sg = h2 + (b * 128 + c + 64) * NT;
  for (int idx = threadIdx.x; idx < 208 * 32; idx += 256) {
    int k = idx >> 5, t = idx & 31;
    bool ok = (k < N_ && t < T_);
    Bf[idx] = ok ? sf[k * T_ + t] : 0.f;
    Bg[idx] = ok ? sg[k * T_ + t] : 0.f;
  }
  __syncthreads();
  int lane = threadIdx.x & 31, wave = threadIdx.x >> 5;
  int lrow = lane & 15, khalf = (lane >> 4) * 2, crow = (lane >> 4) * 8;
  for (int tau = wave; tau < 26; tau += 8) {
    int m0 = (tau >> 1) * 16, n0 = (tau & 1) * 16;
    v8f accF = {}, accG = {};
    for (int k0 = 0; k0 < 208; k0 += 4) {
      int nrow = m0 + lrow, ka = k0 + khalf, kb = ka + 1;
      v2f a, bf, bg;
      a.x = (nrow < N_ && ka < N_) ? adj[nrow * N_ + ka] : 0.f;
      a.y = (nrow < N_ && kb < N_) ? adj[nrow * N_ + kb] : 0.f;
      bf.x = Bf[ka * 32 + n0 + lrow];
      bf.y = Bf[kb * 32 + n0 + lrow];
      bg.x = Bg[ka * 32 + n0 + lrow];
      bg.y = Bg[kb * 32 + n0 + lrow];
      accF = wmma4(a, bf, accF);
      accG = wmma4(a, bg, accG);
    }
    for (int r = 0; r < 8; r++) {
      int row = m0 + crow + r, col = n0 + lrow;
      if (row < N_ && col < T_)
        x1[(b * 64 + c) * NT + row * T_ + col] = sigm(accG[r]) * lrelu(accF[r]);
    }
  }
}

// ---------------- K6: attention features f1 (b,l,n) and f2 (b,c,l) -------------
__global__ __launch_bounds__(256) void k_f1(const float* __restrict__ x1,
                                            const float* __restrict__ w1c,
                                            float* __restrict__ f1) {
  int idx = blockIdx.x * 256 + threadIdx.x;
  if (idx >= B_ * T_ * N_) return;
  int b = idx / (T_ * N_), rem = idx % (T_ * N_), l = rem / N_, n = rem % N_;
  const float* base = x1 + b * PB + n * T_ + l;
  float s = 0.f;
  for (int c = 0; c < 64; c++) s += base[c * NT] * w1c[c];
  f1[idx] = s;
}

__global__ __launch_bounds__(256) void k_f2(const float* __restrict__ x1,
                                            const float* __restrict__ w2c,
                                            float* __restrict__ f2) {
  int idx = blockIdx.x * 256 + threadIdx.x;
  if (idx >= B_ * 64 * T_) return;
  int b = idx / (64 * T_), rem = idx % (64 * T_), c = rem / T_, l = rem % T_;
  const float* base = x1 + (b * 64 + c) * NT + l;
  float s = 0.f;
  for (int n = 0; n < N_; n++) s += base[n * T_] * w2c[n];
  f2[idx] = s;
}

// ------- K7: logits = v_att @ sigmoid(f1 @ w_att @ f2 + b_att), per batch ------
__global__ __launch_bounds__(256) void k_att(const float* __restrict__ f1,
                                             const float* __restrict__ f2,
                                             const float* __restrict__ w_att,
                                             const float* __restrict__ b_att,
                                             const float* __restrict__ v_att,
                                             float* __restrict__ att) {
  int b = blockIdx.x;
  __shared__ float sG[24 * 64];
  __shared__ float sS[24 * 24];
  __shared__ float sF2[64 * 24];
  for (int idx = threadIdx.x; idx < 64 * 24; idx += 256)
    sF2[idx] = f2[b * 64 * 24 + idx];
  for (int idx = threadIdx.x; idx < 24 * 64; idx += 256) {
    int l = idx >> 6, c = idx & 63;
    const float* fr = f1 + (b * 24 + l) * N_;
    float s = 0.f;
    for (int n = 0; n < N_; n++) s += fr[n] * w_att[n * 64 + c];
    sG[idx] = s;
  }
  __syncthreads();
  for (int idx = threadIdx.x; idx < 576; idx += 256) {
    int l = idx / 24, q = idx % 24;
    float s = 0.f;
    for (int c = 0; c < 64; c++) s += sG[l * 64 + c] * sF2[c * 24 + q];
    sS[idx] = sigm(s + b_att[idx]);
  }
  __syncthreads();
  for (int idx = threadIdx.x; idx < 576; idx += 256) {
    int l = idx / 24, q = idx % 24;
    float s = 0.f;
    for (int k = 0; k < 24; k++) s += v_att[l * 24 + k] * sS[k * 24 + q];
    att[b * 576 + idx] = s;
  }
}

// ---------------- K8: BN statistics over (b,l) per q ---------------------------
__global__ __launch_bounds__(256) void k_bnstats(const float* __restrict__ att,
                                                 float* __restrict__ stats) {
  int q = blockIdx.x;
  __shared__ float sS[256], sQ[256];
  float s = 0.f, sq = 0.f;
  for (int i = threadIdx.x; i < 1536; i += 256) {
    float v = att[i * 24 + q];
    s += v; sq += v * v;
  }
  sS[threadIdx.x] = s; sQ[threadIdx.x] = sq;
  __syncthreads();
  for (int o = 128; o > 0; o >>= 1) {
    if (threadIdx.x < o) { sS[threadIdx.x] += sS[threadIdx.x + o]; sQ[threadIdx.x] += sQ[threadIdx.x + o]; }
    __syncthreads();
  }
  if (threadIdx.x == 0) {
    float m = sS[0] / 1536.f;
    float v = sQ[0] / 1536.f - m * m;
    stats[q] = m;
    stats[24 + q] = rsqrtf(v + EPS_BN);
  }
}

// ------- K9: BN apply + softmax over q + transposed store (returned T_coef) ----
__global__ void k_softmax(const float* __restrict__ att, const float* __restrict__ stats,
                          const float* __restrict__ bn_g, const float* __restrict__ bn_b,
                          float* __restrict__ out3) {
  int b = blockIdx.x, l = threadIdx.x;
  if (l >= 24) return;
  float v[24];
  float mx = -1e30f;
  for (int q = 0; q < 24; q++) {
    float z = (att[(b * 24 + l) * 24 + q] - stats[q]) * stats[24 + q] * bn_g[q] + bn_b[q];
    v[q] = z; mx = fmaxf(mx, z);
  }
  float s = 0.f;
  for (int q = 0; q < 24; q++) { v[q] = __expf(v[q] - mx); s += v[q]; }
  float inv = 1.f / s;
  for (int q = 0; q < 24; q++) out3[(b * 24 + q) * 24 + l] = v[q] * inv;
}

// ------- K10: x_att = x1 @ T_ret (per-row 24x24), + lrelu + residual -----------
__global__ __launch_bounds__(256) void k_applyatt(const float* __restrict__ x1,
                                                  const float* __restrict__ tret,
                                                  const float* __restrict__ xin1,
                                                  float* __restrict__ pre) {
  int b = blockIdx.y;
  int row0 = blockIdx.x * 64;
  __shared__ float Bm[24 * 32];
  for (int idx = threadIdx.x; idx < 24 * 32; idx += 256) {
    int l = idx >> 5, q = idx & 31;
    Bm[idx] = (q < 24) ? tret[(b * 24 + l) * 24 + q] : 0.f;
  }
  __syncthreads();
  int lane = threadIdx.x & 31, wave = threadIdx.x >> 5;
  int lrow = lane & 15, khalf = (lane >> 4) * 2, crow = (lane >> 4) * 8;
  int m0 = row0 + (wave >> 1) * 16, n0 = (wave & 1) * 16;
  v8f acc = {};
  for (int k0 = 0; k0 < 24; k0 += 4) {
    v2f a, bb;
    a.x = x1[b * PB + (m0 + lrow) * T_ + k0 + khalf];
    a.y = x1[b * PB + (m0 + lrow) * T_ + k0 + khalf + 1];
    bb.x = Bm[(k0 + khalf) * 32 + n0 + lrow];
    bb.y = Bm[(k0 + khalf + 1) * 32 + n0 + lrow];
    acc = wmma4(a, bb, acc);
  }
  for (int r = 0; r < 8; r++) {
    int row = m0 + crow + r, q = n0 + lrow;
    if (q < T_) {
      int off = b * PB + row * T_ + q;
      pre[off] = lrelu(acc[r]) + xin1[off];
    }
  }
}

// ---------------- K11: LayerNorm (per batch over c,n,t) ------------------------
__global__ __launch_bounds__(256) void k_lnstats(const float* __restrict__ pre,
                                                 float* __restrict__ lst) {
  int b = blockIdx.x;
  __shared__ float sS[256], sQ[256];
  const float* p = pre + (size_t)b * PB;
  float s = 0.f, sq = 0.f;
  for (int i = threadIdx.x; i < PB; i += 256) {
    float v = p[i];
    s += v; sq += v * v;
  }
  sS[threadIdx.x] = s; sQ[threadIdx.x] = sq;
  __syncthreads();
  for (int o = 128; o > 0; o >>= 1) {
    if (threadIdx.x < o) { sS[threadIdx.x] += sS[threadIdx.x + o]; sQ[threadIdx.x] += sQ[threadIdx.x + o]; }
    __syncthreads();
  }
  if (threadIdx.x == 0) {
    float m = sS[0] / (float)PB;
    float v = sQ[0] / (float)PB - m * m;
    lst[b] = m;
    lst[64 + b] = rsqrtf(v + EPS_LN);
  }
}

__global__ __launch_bounds__(256) void k_lnapply(const float* __restrict__ pre,
                                                 const float* __restrict__ lst,
                                                 const float* __restrict__ ln_g,
                                                 const float* __restrict__ ln_b,
                                                 float* __restrict__ out) {
  int idx = blockIdx.x * 256 + threadIdx.x;
  if (idx >= S1) return;
  int b = idx / PB, r = idx % PB;
  out[idx] = (pre[idx] - lst[b]) * lst[64 + b] * ln_g[r] + ln_b[r];
}

extern "C" void kernel_launch(void* const* d_in, const int* in_sizes, int n_in,
                              void* d_out, int out_size, void* d_ws, size_t ws_size,
                              hipStream_t stream) {
  const float* x    = (const float*)d_in[0];
  const float* adj  = (const float*)d_in[2];   // d_in[1] = supports (unused by reference)
  const float* Wt   = (const float*)d_in[3];
  const float* bt   = (const float*)d_in[4];
  const float* W2   = (const float*)d_in[5];
  const float* b2   = (const float*)d_in[6];
  const float* W1   = (const float*)d_in[7];
  const float* b1   = (const float*)d_in[8];
  const float* w1c  = (const float*)d_in[9];
  const float* w2c  = (const float*)d_in[10];
  const float* watt = (const float*)d_in[11];
  const float* batt = (const float*)d_in[12];
  const float* vatt = (const float*)d_in[13];
  const float* bng  = (const float*)d_in[14];
  const float* bnb  = (const float*)d_in[15];
  const float* lng  = (const float*)d_in[16];
  const float* lnb  = (const float*)d_in[17];

  float* ws   = (float*)d_ws;
  float* xin1 = ws;                       // buf A, live to the end (residual)
  float* xin2 = ws + (size_t)S1;          // buf B
  float* h2   = ws + (size_t)2 * S1;      // buf C (2*S1): t1, then h2, then pre
  float* t1   = h2;
  float* x1   = xin2;                     // reuse B after K4
  float* pre  = h2;                       // reuse C after K5
  float* f1   = ws + (size_t)4 * S1;
  float* f2   = f1 + (size_t)B_ * T_ * N_;
  float* att  = f2 + (size_t)B_ * 64 * T_;
  float* stats = att + (size_t)B_ * T_ * T_;
  float* lst   = stats + 48;

  float* out     = (float*)d_out;
  float* out_adj = out + (size_t)S1;
  float* out3    = out_adj + (size_t)N_ * N_;

  k_conv<<<B_ * N_, 256, 0, stream>>>(x, Wt, bt, xin1);
  dim3 gcm((NT + 31) / 32, B_);
  k_chanmix<<<gcm, 256, 0, stream>>>(xin1, W2, b2, t1, 64);
  k_adjmm<<<B_ * 64, 256, 0, stream>>>(adj, t1, xin2);
  k_chanmix<<<gcm, 256, 0, stream>>>(xin2, W1, b1, h2, 128);
  k_adjmm_gated<<<B_ * 64, 256, 0, stream>>>(adj, h2, x1);
  k_f1<<<(B_ * T_ * N_ + 255) / 256, 256, 0, stream>>>(x1, w1c, f1);
  k_f2<<<(B_ * 64 * T_ + 255) / 256, 256, 0, stream>>>(x1, w2c, f2);
  k_att<<<B_, 256, 0, stream>>>(f1, f2, watt, batt, vatt, att);
  k_bnstats<<<24, 256, 0, stream>>>(att, stats);
  k_softmax<<<B_, 32, 0, stream>>>(att, stats, bng, bnb, out3);
  dim3 gap(207, B_);
  k_applyatt<<<gap, 256, 0, stream>>>(x1, out3, xin1, pre);
  k_lnstats<<<B_, 256, 0, stream>>>(pre, lst);
  k_lnapply<<<(S1 + 255) / 256, 256, 0, stream>>>(pre, lst, lng, lnb, out);
  hipMemcpyAsync(out_adj, adj, (size_t)N_ * N_ * sizeof(float),
                 hipMemcpyDeviceToDevice, stream);
}